// PositionalEncoding_5987184411234
// MI455X (gfx1250) — compile-verified
//
#include <hip/hip_runtime.h>
#include <stdint.h>

// Problem constants (match reference)
#define BATCH 16
#define SEQ   2048
#define DIM   2048

typedef float v4f __attribute__((ext_vector_type(4)));

// ---------------------------------------------------------------------------
// Kernel 1: verse weights via wave32 segmented scan.
// One block (= one wave of 32 lanes) per batch row. Lane l owns the
// contiguous chunk v[l*64 .. l*64+63].
//   pass 1: per-lane (lastNonzeroVal, hasNonzero)
//   scan  : Hillis-Steele over lanes, combine(a,b) = b.has ? b : a
//   pass 2: replay chunk with incoming 'prev' to emit 1.2 / 1.0
// ---------------------------------------------------------------------------
__global__ void __launch_bounds__(32)
verse_weights_kernel(const int* __restrict__ v, float* __restrict__ w) {
    const int row  = blockIdx.x;          // 0..BATCH-1
    const int lane = threadIdx.x;         // 0..31  (wave32)
    const int* vr  = v + row * SEQ;
    float*     wr  = w + row * SEQ;
    const int  base = lane * (SEQ / 32);  // 64 elements per lane

    // pass 1: chunk summary
    int lastVal = 0, has = 0;
    #pragma unroll 8
    for (int k = 0; k < SEQ / 32; ++k) {
        int val = vr[base + k];
        if (val != 0) { lastVal = val; has = 1; }
    }

    // inclusive scan across the wave (wave32)
    int sVal = lastVal, sHas = has;
    #pragma unroll
    for (int off = 1; off < 32; off <<= 1) {
        int oVal = __shfl_up(sVal, off, 32);
        int oHas = __shfl_up(sHas, off, 32);
        if (lane >= off && !sHas) { sVal = oVal; sHas = oHas; }
    }
    // exclusive prefix -> value of the last nonzero verse before my chunk
    int pVal = __shfl_up(sVal, 1, 32);
    int pHas = __shfl_up(sHas, 1, 32);
    int prev = (lane > 0 && pHas) ? pVal : -1;   // -1 == "no previous verse"

    // pass 2: emit weights (chunk re-read hits L0/L2, it's tiny)
    #pragma unroll 8
    for (int k = 0; k < SEQ / 32; ++k) {
        int val = vr[base + k];
        float wt = (val != 0 && val != prev) ? 1.2f : 1.0f;
        if (val != 0) prev = val;
        wr[base + k] = wt;
    }
}

// ---------------------------------------------------------------------------
// Kernel 2: out[b,s,:] = x[b,s,:] + pe[s,:] * w[b,s]
// One block per sequence position s. pe row (8 KB) is staged into LDS with
// gfx1250 async global->LDS b128 loads, then held in VGPRs across the batch
// loop. x/out are streaming (touch-once) -> non-temporal hints so they don't
// evict pe's 16 MB working set from the 192 MB L2.
// ---------------------------------------------------------------------------
__global__ void __launch_bounds__(256)
add_pe_kernel(const float* __restrict__ x,
              const float* __restrict__ pe,
              const float* __restrict__ w,
              float* __restrict__ out) {
    __shared__ float spe[DIM];            // 8 KB

    const int s = blockIdx.x;             // 0..SEQ-1
    const int t = threadIdx.x;            // 0..255

    // ---- async copy pe[s, 0:2048] -> LDS (512 x b128, 2 per thread) ----
    const char* src = (const char*)(pe + (size_t)s * DIM);
    #pragma unroll
    for (int i = 0; i < 2; ++i) {
        const int idx4 = t + i * 256;                       // float4 index
        unsigned lds_addr = (unsigned)(uintptr_t)(&spe[idx4 * 4]);
        uint64_t gaddr    = (uint64_t)(src + (size_t)idx4 * 16);
        asm volatile("global_load_async_to_lds_b128 %0, %1, off"
                     :: "v"(lds_addr), "v"(gaddr)
                     : "memory");
    }
    asm volatile("s_wait_asynccnt 0" ::: "memory");   // my wave's asyncs done
    __syncthreads();                                  // all waves' asyncs done

    // ---- pull my 8 pe values into VGPRs once ----
    const v4f p0 = *(const v4f*)&spe[t * 8];
    const v4f p1 = *(const v4f*)&spe[t * 8 + 4];

    const size_t rowOff = (size_t)s * DIM + (size_t)t * 8;

    #pragma unroll
    for (int b = 0; b < BATCH; ++b) {
        const float wgt = w[b * SEQ + s];             // uniform -> scalar load
        const v4f* xp = (const v4f*)(x   + (size_t)b * SEQ * DIM + rowOff);
        v4f*       op = (v4f*)      (out + (size_t)b * SEQ * DIM + rowOff);

        v4f x0 = __builtin_nontemporal_load(xp);
        v4f x1 = __builtin_nontemporal_load(xp + 1);

        v4f r0 = x0 + p0 * wgt;
        v4f r1 = x1 + p1 * wgt;

        __builtin_nontemporal_store(r0, op);
        __builtin_nontemporal_store(r1, op + 1);
    }
}

// ---------------------------------------------------------------------------
extern "C" void kernel_launch(void* const* d_in, const int* in_sizes, int n_in,
                              void* d_out, int out_size, void* d_ws, size_t ws_size,
                              hipStream_t stream) {
    const float* x  = (const float*)d_in[0];   // [16,2048,2048] fp32
    const float* pe = (const float*)d_in[1];   // [2048,2048]    fp32
    const int*   vp = (const int*)  d_in[2];   // [16,2048]      int32
    float*       out = (float*)d_out;          // [16,2048,2048] fp32
    float*       wts = (float*)d_ws;           // [16,2048] fp32 scratch (128 KB)

    verse_weights_kernel<<<BATCH, 32, 0, stream>>>(vp, wts);
    add_pe_kernel<<<SEQ, 256, 0, stream>>>(x, pe, wts, out);
}